// PointDeformation_59347858096825
// MI455X (gfx1250) — compile-verified
//
#include <hip/hip_runtime.h>
#include <math.h>

typedef __attribute__((ext_vector_type(2))) float v2f;
typedef __attribute__((ext_vector_type(8))) float v8f;

#define D_MODEL 256
#define T_LEN   2048
#define N_BATCH 32
#define NR_ROWS 48
#define QPTS    32
#define NBDY    4
#define NROWS   (N_BATCH * NR_ROWS)   // 1536

// ---------------------------------------------------------------------------
// Kernel 1: lin[row, 0:16] = pro[row, :] @ Wcat^T + bcat
// Wcat rows 0-3 = w_bdy_off, 4-7 = w_ctr_off, 8-11 = w_bdy_wt, 12-15 = w_ctr_wt
// One wave per 16-row tile; 64 x v_wmma_f32_16x16x4_f32 along K=256.
// ---------------------------------------------------------------------------
__global__ __launch_bounds__(32) void pd_linear_wmma(
    const float* __restrict__ pro,
    const float* __restrict__ w_bdy_off, const float* __restrict__ b_bdy_off,
    const float* __restrict__ w_ctr_off, const float* __restrict__ b_ctr_off,
    const float* __restrict__ w_bdy_wt,  const float* __restrict__ b_bdy_wt,
    const float* __restrict__ w_ctr_wt,  const float* __restrict__ b_ctr_wt,
    float* __restrict__ lin)
{
    const int lane = threadIdx.x;
    const int m    = lane & 15;   // A: row M; B: col N; D: col N
    const int hi   = lane >> 4;   // K-half selector for A/B fragments

    const float* Arow = pro + (size_t)(blockIdx.x * 16 + m) * D_MODEL;

    // Select which concatenated weight row this lane's B fragment comes from.
    const int g = m >> 2;
    const float* Wsel = (g == 0) ? w_bdy_off : (g == 1) ? w_ctr_off
                      : (g == 2) ? w_bdy_wt  : w_ctr_wt;
    const float* Brow = Wsel + (size_t)(m & 3) * D_MODEL;
    const float* Bsel = (g == 0) ? b_bdy_off : (g == 1) ? b_ctr_off
                      : (g == 2) ? b_bdy_wt  : b_ctr_wt;
    const float bias = Bsel[m & 3];

    v8f acc = {};
    #pragma unroll 4
    for (int k = 0; k < D_MODEL; k += 4) {
        // A 16x4 f32 layout: lanes 0-15 hold K=k..k+1, lanes 16-31 hold K=k+2..k+3
        v2f a = *(const v2f*)(Arow + k + 2 * hi);
        // B 4x16 f32 layout mirrors A with N across lanes
        v2f b = *(const v2f*)(Brow + k + 2 * hi);
        acc = __builtin_amdgcn_wmma_f32_16x16x4_f32(
            /*neg_a=*/false, a, /*neg_b=*/false, b,
            /*c_mod=*/(short)0, acc, /*reuse_a=*/false, /*reuse_b=*/false);
    }

    // D layout: VGPR j holds M = j (lanes 0-15) / M = j+8 (lanes 16-31), N = lane%16
    float* outt = lin + (size_t)blockIdx.x * 16 * 16;
    #pragma unroll
    for (int j = 0; j < 8; ++j) {
        const int mrow = j + 8 * hi;
        outt[mrow * 16 + m] = acc[j] + bias;
    }
}

// ---------------------------------------------------------------------------
// Kernel 2: per (n, r) block, 256 threads = 256 channels.
// Softmax the mixing weights, then for each of 32 query points accumulate
// 4 bilinear subpoint samples. All feature loads are coalesced 1 KB rows;
// all address math is block-uniform (scalarized by the compiler).
// ---------------------------------------------------------------------------
__global__ __launch_bounds__(256) void pd_sample(
    const float* __restrict__ features,
    const float* __restrict__ sampled,
    const float* __restrict__ lin,
    float* __restrict__ out)
{
    const int r   = blockIdx.x;            // 0..47
    const int n   = blockIdx.y;            // 0..31
    const int row = n * NR_ROWS + r;       // 0..1535
    const int d   = threadIdx.x;           // channel

    const float* L = lin + (size_t)row * 16;
    float bo[4], co[4], bw[4], cw[4];
    #pragma unroll
    for (int s = 0; s < 4; ++s) {
        bo[s] = L[s]; co[s] = L[4 + s]; bw[s] = L[8 + s]; cw[s] = L[12 + s];
    }
    // 4-wide softmaxes (uniform across block)
    float mB = fmaxf(fmaxf(bw[0], bw[1]), fmaxf(bw[2], bw[3]));
    float mC = fmaxf(fmaxf(cw[0], cw[1]), fmaxf(cw[2], cw[3]));
    float sB = 0.f, sC = 0.f;
    #pragma unroll
    for (int s = 0; s < 4; ++s) {
        bw[s] = expf(bw[s] - mB); sB += bw[s];
        cw[s] = expf(cw[s] - mC); sC += cw[s];
    }
    const float rB = 1.f / sB, rC = 1.f / sC;
    #pragma unroll
    for (int s = 0; s < 4; ++s) { bw[s] *= rB; cw[s] *= rC; }

    const float* feat = features + (size_t)n * T_LEN * D_MODEL + d;
    const float* sp   = sampled  + (size_t)row * QPTS;
    float* outp       = out      + (size_t)row * D_MODEL + d;

    #pragma unroll 2
    for (int q = 0; q < QPTS; ++q) {
        const float p   = sp[q];
        const bool isB  = (q < NBDY);
        float acc = 0.f;
        #pragma unroll
        for (int s = 0; s < 4; ++s) {
            const float off = isB ? bo[s] : co[s];
            const float wt  = isB ? bw[s] : cw[s];
            float loc = p + off * (1.0f / 2048.0f);
            loc = fminf(fmaxf(loc, 0.0f), 1.0f);
            const float px  = loc * 2048.0f - 0.5f;
            const float x0f = floorf(px);
            const float w1  = px - x0f;
            const int   x0  = (int)x0f;                 // in [-1, 2047]
            const float f0 = (x0 >= 0)
                ? feat[(size_t)x0 * D_MODEL] : 0.0f;    // x0 <= 2047 guaranteed
            const float f1 = (x0 < T_LEN - 1)
                ? feat[(size_t)(x0 + 1) * D_MODEL] : 0.0f;
            acc += wt * 0.5f * ((1.0f - w1) * f0 + w1 * f1);
        }
        outp[(size_t)q * NROWS * D_MODEL] = acc;
    }
}

// ---------------------------------------------------------------------------
extern "C" void kernel_launch(void* const* d_in, const int* in_sizes, int n_in,
                              void* d_out, int out_size, void* d_ws, size_t ws_size,
                              hipStream_t stream) {
    const float* pro       = (const float*)d_in[0];
    const float* features  = (const float*)d_in[1];
    const float* sampled   = (const float*)d_in[2];
    const float* w_bdy_off = (const float*)d_in[3];
    const float* b_bdy_off = (const float*)d_in[4];
    const float* w_ctr_off = (const float*)d_in[5];
    const float* b_ctr_off = (const float*)d_in[6];
    const float* w_bdy_wt  = (const float*)d_in[7];
    const float* b_bdy_wt  = (const float*)d_in[8];
    const float* w_ctr_wt  = (const float*)d_in[9];
    const float* w_ctr_b   = (const float*)d_in[10];

    float* lin = (float*)d_ws;   // 1536 x 16 floats = 96 KB scratch

    pd_linear_wmma<<<NROWS / 16, 32, 0, stream>>>(
        pro, w_bdy_off, b_bdy_off, w_ctr_off, b_ctr_off,
        w_bdy_wt, b_bdy_wt, w_ctr_wt, w_ctr_b, lin);

    dim3 grid2(NR_ROWS, N_BATCH);
    pd_sample<<<grid2, 256, 0, stream>>>(features, sampled, lin, (float*)d_out);
}